// SSD_74646531604922
// MI455X (gfx1250) — compile-verified
//
#include <hip/hip_runtime.h>
#include <math.h>

// Problem constants (match reference)
#define B_      32
#define P_      8732
#define C_      21
#define TOP_K_  200
#define PRE_    400
#define CONF_T_ 0.01f
#define NMS_T_  0.45f

#define NSORT    16384          // next pow2 >= P_
#define NTHREADS 256            // 8 waves of 32
#define NTILE    25             // PRE_/16
#define WPR      16             // u32 words per suppression row (>= ceil(400/32)=13)

typedef __attribute__((ext_vector_type(2))) float v2f;
typedef __attribute__((ext_vector_type(8))) float v8f;

// Order-preserving float <-> uint mapping (ascending uint == ascending float)
__device__ __forceinline__ unsigned ord_f32(float f) {
    unsigned u = __float_as_uint(f);
    return (u & 0x80000000u) ? ~u : (u | 0x80000000u);
}
__device__ __forceinline__ float unord_f32(unsigned o) {
    unsigned u = (o & 0x80000000u) ? (o & 0x7FFFFFFFu) : ~o;
    return __uint_as_float(u);
}

// ---------------------------------------------------------------------------
// Kernel 1: per-(b,p) log-sum-exp over the C=21 class logits
// ---------------------------------------------------------------------------
__global__ void lse_kernel(const float* __restrict__ conf,
                           float* __restrict__ lse, int n) {
    int i = blockIdx.x * blockDim.x + threadIdx.x;
    if (i >= n) return;
    const float* row = conf + (size_t)i * C_;
    float m = row[0];
#pragma unroll
    for (int c = 1; c < C_; ++c) m = fmaxf(m, row[c]);
    float s = 0.f;
#pragma unroll
    for (int c = 0; c < C_; ++c) s += expf(row[c] - m);
    lse[i] = m + logf(s);
}

// ---------------------------------------------------------------------------
// Kernel 2: one workgroup per (batch, foreground class).
// ---------------------------------------------------------------------------
__global__ __launch_bounds__(NTHREADS) void nms_kernel(
    const float* __restrict__ loc, const float* __restrict__ conf,
    const float* __restrict__ priors, const float* __restrict__ lse,
    float* __restrict__ out) {
    // 128 KB of LDS: sort buffer, later overlaid with candidate/NMS state.
    __shared__ unsigned long long keys[NSORT];

    const int tid = threadIdx.x;
    const int b = blockIdx.x / (C_ - 1);
    const int c = blockIdx.x % (C_ - 1) + 1;   // skip background class 0

    const float* confb = conf + (size_t)b * P_ * C_ + c;
    const float* lseb  = lse + (size_t)b * P_;

    // ---- Phase 1: build sortable keys: [ordered(score) | ~prior_index] ----
    for (int p = tid; p < NSORT; p += NTHREADS) {
        float s = -2.0f;                       // pad sentinel
        if (p < P_) {
            if (p + 4 * NTHREADS < P_)
                __builtin_prefetch(&confb[(size_t)(p + 4 * NTHREADS) * C_], 0, 1);
            float sc = expf(confb[(size_t)p * C_] - lseb[p]);
            s = (sc > CONF_T_) ? sc : -1.0f;   // reference: where(s>T, s, -1)
        }
        keys[p] = ((unsigned long long)ord_f32(s) << 32) |
                  (unsigned long long)(~(unsigned)p);  // ties -> smaller p first
    }
    __syncthreads();

    // ---- Phase 2: bitonic sort ascending (top candidates end up last) ----
    for (unsigned k = 2; k <= NSORT; k <<= 1) {
        for (unsigned j = k >> 1; j > 0; j >>= 1) {
            for (unsigned i = tid; i < NSORT; i += NTHREADS) {
                unsigned ixj = i ^ j;
                if (ixj > i) {
                    unsigned long long a = keys[i], bb = keys[ixj];
                    bool asc = ((i & k) == 0);
                    if (asc ? (a > bb) : (a < bb)) { keys[i] = bb; keys[ixj] = a; }
                }
            }
            __syncthreads();
        }
    }

    // ---- Overlay candidate/NMS state onto low part of the sort buffer ----
    char* base = (char*)keys;
    float* s_score = (float*)(base + 4096);
    float* s_x1 = s_score + PRE_;
    float* s_y1 = s_x1 + PRE_;
    float* s_x2 = s_y1 + PRE_;
    float* s_y2 = s_x2 + PRE_;
    float* s_area = s_y2 + PRE_;
    unsigned* s_sup = (unsigned*)(s_area + PRE_);            // PRE_*WPR words
    volatile unsigned* s_keep = (volatile unsigned*)(s_sup + PRE_ * WPR);
    unsigned* s_wpre = (unsigned*)(s_sup + PRE_ * WPR + WPR);

    // ---- Phase 3: extract top-400, decode SSD boxes, areas ----
    for (int t = tid; t < PRE_; t += NTHREADS) {
        unsigned long long key = keys[NSORT - 1 - t];
        float sc = unord_f32((unsigned)(key >> 32));
        unsigned p = ~(unsigned)(key & 0xFFFFFFFFu);
        float x1 = 0.f, y1 = 0.f, x2 = 0.f, y2 = 0.f;
        if (p < P_) {
            const float* lp = loc + ((size_t)b * P_ + p) * 4;
            const float* pp = priors + (size_t)p * 4;
            float cx = pp[0] + lp[0] * 0.1f * pp[2];
            float cy = pp[1] + lp[1] * 0.1f * pp[3];
            float w  = pp[2] * expf(lp[2] * 0.2f);
            float h  = pp[3] * expf(lp[3] * 0.2f);
            x1 = cx - 0.5f * w; y1 = cy - 0.5f * h;
            x2 = cx + 0.5f * w; y2 = cy + 0.5f * h;
        }
        s_score[t] = sc;
        s_x1[t] = x1; s_y1[t] = y1; s_x2[t] = x2; s_y2[t] = y2;
        s_area[t] = (x2 - x1) * (y2 - y1);
    }
    for (int i = tid; i < PRE_ * WPR; i += NTHREADS) s_sup[i] = 0;
    __syncthreads();

    // ---- Phase 4: IoU tiles; WMMA computes (area_i + area_j) per 16x16 tile
    // Division-free threshold: inter/den > T  <=>  inter > T*den  (den > 0).
    const int lane = tid & 31;
    const int wid  = tid >> 5;
    const int half = lane >> 4;     // 0: lanes 0-15, 1: lanes 16-31
    const int l16  = lane & 15;
    for (int tile = wid; tile < NTILE * NTILE; tile += NTHREADS / 32) {
        int ti = tile / NTILE, tj = tile % NTILE;
        v2f A, Bv;
        A.x  = half ? 0.f : s_area[ti * 16 + l16];   // K=0 (lanes 0-15)
        A.y  = half ? 0.f : 1.0f;                    // K=1
        Bv.x = half ? 0.f : 1.0f;                    // K=0 row of ones
        Bv.y = half ? 0.f : s_area[tj * 16 + l16];   // K=1 row = col areas
        v8f D = {0.f, 0.f, 0.f, 0.f, 0.f, 0.f, 0.f, 0.f};
        D = __builtin_amdgcn_wmma_f32_16x16x4_f32(
            /*neg_a=*/false, A, /*neg_b=*/false, Bv,
            /*c_mod=*/(short)0, D, /*reuse_a=*/false, /*reuse_b=*/false);

        int n = tj * 16 + l16;
        float cx1 = s_x1[n], cy1 = s_y1[n], cx2 = s_x2[n], cy2 = s_y2[n];
        unsigned myword = (unsigned)(tj >> 1);
        unsigned shiftb = (unsigned)((tj & 1) * 16);
#pragma unroll
        for (int v = 0; v < 8; ++v) {
            int m = ti * 16 + v + 8 * half;          // C/D layout: row = v (+8 hi half)
            float rx1 = s_x1[m], ry1 = s_y1[m], rx2 = s_x2[m], ry2 = s_y2[m];
            float iw = fmaxf(fminf(rx2, cx2) - fmaxf(rx1, cx1), 0.f);
            float ih = fmaxf(fminf(ry2, cy2) - fmaxf(ry1, cy1), 0.f);
            float inter = iw * ih;
            float den = D[v] - inter + 1e-12f;       // D[v] = area_m + area_n
            bool sup = (inter > NMS_T_ * den) && (n > m);
            unsigned long long bal = __ballot(sup);
            if (l16 == 0) {    // lane 0 emits rows of low half, lane 16 high half
                unsigned bits = half ? (unsigned)((bal >> 16) & 0xFFFFu)
                                     : (unsigned)(bal & 0xFFFFu);
                if (bits) atomicOr(&s_sup[m * WPR + myword], bits << shiftb);
            }
        }
    }
    // keep-mask init: valid = score > CONF_T
    if (tid < WPR) {
        unsigned w = 0;
        for (int jj = 0; jj < 32; ++jj) {
            int j = tid * 32 + jj;
            if (j < PRE_ && s_score[j] > CONF_T_) w |= (1u << jj);
        }
        s_keep[tid] = w;
    }
    __syncthreads();

    // ---- Phase 5: greedy suppression scan (one wave, bitmask) ----
    if (tid < 32) {
        for (int i = 0; i < PRE_; ++i) {
            unsigned kw = s_keep[i >> 5];
            if ((kw >> (i & 31)) & 1u) {
                if (lane < WPR)
                    s_keep[lane] = s_keep[lane] & ~s_sup[i * WPR + lane];
            }
        }
    }
    __syncthreads();
    if (tid == 0) {
        unsigned run = 0;
        for (int w = 0; w < WPR; ++w) { s_wpre[w] = run; run += __popc(s_keep[w]); }
    }
    __syncthreads();

    // ---- Phase 6: compact kept rows into out[b][c][rank] (rank < TOP_K) ----
    float* outbc = out + ((size_t)b * C_ + c) * TOP_K_ * 5;
    for (int t = tid; t < PRE_; t += NTHREADS) {
        unsigned word = s_keep[t >> 5];
        if ((word >> (t & 31)) & 1u) {
            unsigned rank = s_wpre[t >> 5] + __popc(word & ((1u << (t & 31)) - 1u));
            if (rank < TOP_K_) {
                float* r = outbc + (size_t)rank * 5;
                r[0] = s_score[t];
                r[1] = s_x1[t]; r[2] = s_y1[t]; r[3] = s_x2[t]; r[4] = s_y2[t];
            }
        }
    }
}

// ---------------------------------------------------------------------------
extern "C" void kernel_launch(void* const* d_in, const int* in_sizes, int n_in,
                              void* d_out, int out_size, void* d_ws, size_t ws_size,
                              hipStream_t stream) {
    const float* loc    = (const float*)d_in[0];  // [B,P,4]
    const float* conf   = (const float*)d_in[1];  // [B,P,C]
    const float* priors = (const float*)d_in[2];  // [P,4]
    float* out = (float*)d_out;                   // [B,C,TOP_K,5]
    float* lse = (float*)d_ws;                    // [B*P] floats

    // Output is mostly zeros (background class + padding): clear it first.
    hipMemsetAsync(d_out, 0, (size_t)out_size * sizeof(float), stream);

    int n = B_ * P_;
    lse_kernel<<<(n + 255) / 256, 256, 0, stream>>>(conf, lse, n);
    nms_kernel<<<B_ * (C_ - 1), NTHREADS, 0, stream>>>(loc, conf, priors, lse, out);
}